// AttentionModule_1924145349105
// MI455X (gfx1250) — compile-verified
//
#include <hip/hip_runtime.h>
#include <math.h>

typedef __attribute__((ext_vector_type(2))) float v2f;
typedef __attribute__((ext_vector_type(8))) float v8f;

#define HDIM 64

// ---- orderable-float helpers: monotonic bijection f32 -> u32 -----------------
__device__ __forceinline__ unsigned ford(float x) {
  unsigned u = __float_as_uint(x);
  return (u & 0x80000000u) ? ~u : (u | 0x80000000u);
}
__device__ __forceinline__ float finv(unsigned o) {
  unsigned u = (o & 0x80000000u) ? (o & 0x7FFFFFFFu) : ~o;
  return __uint_as_float(u);
}
__device__ __forceinline__ int frombin4(const int* b) {
  return b[0] + 2 * b[1] + 4 * b[2] + 8 * b[3];
}

// ---- Kernel 1: all seven lookup tables via V_WMMA_F32_16X16X4_F32 ------------
// tabs layout (floats): Qtab[0..2048) Ktab[2048..4096) Vtab[4096..6144)
//                       EKtab[6144..8192) T1[8192..9216) T2[9216..10240) T3[10240..11264)
__global__ __launch_bounds__(32) void tables_wmma(
    const float* __restrict__ emb_edge, const float* __restrict__ emb_sbv,
    const float* __restrict__ emb_sbr, const float* __restrict__ emb_static,
    const float* __restrict__ Wq, const float* __restrict__ Wk,
    const float* __restrict__ Wv, const float* __restrict__ Wek,
    const float* __restrict__ Wc, float* __restrict__ tabs) {
  const int bid = blockIdx.x;
  const int lane = threadIdx.x;
  const float* A;
  const float* W;
  float* out;
  int arows, ldw, koff, mtile, ntile;
  if (bid < 32) {                       // Qtab/Ktab/Vtab/EKtab: 32x64 = 2x4 tiles each
    int grp = bid >> 3, sub = bid & 7;
    mtile = sub >> 2; ntile = sub & 3;
    arows = 32; ldw = 64; koff = 0;
    A = (grp == 3) ? emb_sbr : emb_sbv;
    W = (grp == 0) ? Wq : (grp == 1) ? Wk : (grp == 2) ? Wv : Wek;
    out = tabs + grp * 2048;
  } else {                              // T1/T2/T3: 16x64 (T3 zero-padded to 16 rows)
    int j = bid - 32;
    int grp = j >> 2;
    mtile = 0; ntile = j & 3;
    A = (grp == 2) ? emb_static : emb_edge;
    arows = (grp == 2) ? 4 : 16;
    W = Wc; ldw = 192; koff = grp * 64;
    out = tabs + 8192 + grp * 1024;
  }
  const int m = mtile * 16 + (lane & 15);    // A row per ISA layout
  const int n = ntile * 16 + (lane & 15);    // B col per ISA layout
  const int kh = (lane >> 4) * 2;            // K-pair select per lane half
  const bool mv = (m < arows);
  v8f acc = {};
  for (int kk = 0; kk < 16; ++kk) {          // K = 64 in steps of 4
    const int kb = kk * 4 + kh;
    v2f a, b;
    a.x = mv ? A[m * 64 + kb] : 0.0f;
    a.y = mv ? A[m * 64 + kb + 1] : 0.0f;
    b.x = W[n * ldw + koff + kb];            // B[k][n] = W[n][koff+k]  (out = A @ W.T)
    b.y = W[n * ldw + koff + kb + 1];
    acc = __builtin_amdgcn_wmma_f32_16x16x4_f32(
        false, a, false, b, (short)0, acc, false, false);
  }
  const int rb = mtile * 16 + (lane >> 4) * 8;  // D: VGPR j -> row j (+8 for hi half)
  const int c = ntile * 16 + (lane & 15);
#pragma unroll
  for (int j = 0; j < 8; ++j) out[(rb + j) * 64 + c] = acc[j];
}

// ---- Kernel 2: node-level select_best argmin (packed u64 atomicMin) ----------
__global__ void node_argmin(const int* __restrict__ node_states,
                            const float* __restrict__ scalars,
                            const int* __restrict__ batch_vec,
                            unsigned long long* __restrict__ nodeKey, int n) {
  int i = blockIdx.x * blockDim.x + threadIdx.x;
  if (i >= n) return;
  int s = frombin4(node_states + i * 4);
  int key = (2 * s) * 64 + batch_vec[i];  // states*NGRAPH + batch
  unsigned long long packed = ((unsigned long long)ford(scalars[i]) << 32) | (unsigned)i;
  atomicMin(&nodeKey[key], packed);
}

// ---- Kernel 3: nidx + node_out base = emb_sbv[nidx] --------------------------
__global__ void node_fill(const int* __restrict__ node_states,
                          const int* __restrict__ batch_vec,
                          const unsigned long long* __restrict__ nodeKey,
                          const float* __restrict__ emb_sbv,
                          int* __restrict__ nidx, float* __restrict__ node_out, int n) {
  int gid = blockIdx.x * blockDim.x + threadIdx.x;
  int i = gid >> 6, h = gid & 63;
  if (i >= n) return;
  int s = 2 * frombin4(node_states + i * 4);
  int key = s * 64 + batch_vec[i];
  unsigned winner = (unsigned)(nodeKey[key] & 0xFFFFFFFFull);
  int idx = s + (winner == (unsigned)i ? 1 : 0);
  if (h == 0) nidx[i] = idx;
  node_out[(size_t)i * HDIM + h] = emb_sbv[idx * 64 + h];
}

// ---- Kernel 4: edge-level select_best argmin ---------------------------------
__global__ void edge_argmin(const int* __restrict__ node_states,
                            const float* __restrict__ scalars,
                            const int* __restrict__ ei0, const int* __restrict__ ei1,
                            unsigned long long* __restrict__ edgeKey, int eN, int n) {
  int e = blockIdx.x * blockDim.x + threadIdx.x;
  if (e >= eN) return;
  int fb = frombin4(node_states + ei0[e] * 4);
  int key = fb * n + ei1[e];  // even-states halved: fb*N + receiver
  unsigned long long packed = ((unsigned long long)ford(scalars[e]) << 32) | (unsigned)e;
  atomicMin(&edgeKey[key], packed);
}

// ---- Kernel 5: per-edge logits (32 lanes/edge) + aux bits + segment max ------
__global__ __launch_bounds__(256) void edge_logits(
    const int* __restrict__ node_states, const int* __restrict__ edge_states,
    const float* __restrict__ scalars, const int* __restrict__ ei0,
    const int* __restrict__ ei1, const unsigned long long* __restrict__ edgeKey,
    const int* __restrict__ nidx, const float* __restrict__ tabs,
    int* __restrict__ aux, float* __restrict__ logit,
    unsigned* __restrict__ recvMax, int eN, int n) {
  int e = (blockIdx.x * blockDim.x + threadIdx.x) >> 5;
  int lane = threadIdx.x & 31;
  if (e >= eN) return;
  int s0 = ei0[e], s1 = ei1[e];
  int fb0 = frombin4(node_states + s0 * 4);
  unsigned winner = (unsigned)(edgeKey[fb0 * n + s1] & 0xFFFFFFFFull);
  int ekidx = 2 * fb0 + (winner == (unsigned)e ? 1 : 0);
  int eidx = frombin4(edge_states + e * 4);
  float sc = scalars[e];
  float rs = scalars[s1];  // node_scalars = scalars[:N]
  float ss = scalars[s0];
  int sidx = (sc < rs ? 1 : 0) + 2 * ((ss + sc) < rs ? 1 : 0);
  if (lane == 0) aux[e] = eidx | (ekidx << 4) | (sidx << 9);
  const float* Qt = tabs + nidx[s1] * 64;
  const float* Kt = tabs + 2048 + nidx[s0] * 64;
  const float* Et = tabs + 6144 + ekidx * 64;
  float p = Qt[lane] * (Kt[lane] + Et[lane]) +
            Qt[lane + 32] * (Kt[lane + 32] + Et[lane + 32]);
  for (int off = 16; off > 0; off >>= 1) p += __shfl_xor(p, off, 32);
  if (lane == 0) {
    float lg = p * 0.125f;  // 1/sqrt(64)
    logit[e] = lg;
    atomicMax(&recvMax[s1], ford(lg));
  }
}

// ---- Kernel 6: exp((logit-max)/tau) + segment sum ----------------------------
__global__ void edge_exp(const int* __restrict__ ei1, float* __restrict__ logit,
                         const unsigned* __restrict__ recvMax,
                         float* __restrict__ recvSum,
                         const int* __restrict__ tstep, int eN) {
  int e = blockIdx.x * blockDim.x + threadIdx.x;
  if (e >= eN) return;
  int s1 = ei1[e];
  float m = finv(recvMax[s1]);
  int step = tstep[0];
  float tau;
  if (step == -1) {
    tau = 1.0f;  // TEMP_ON_EVAL
  } else {
    float frac = fminf((float)step / 10000.0f, 1.0f);
    tau = 3.0f + (0.1f - 3.0f) * frac;
  }
  float ev = expf((logit[e] - m) / tau);
  logit[e] = ev;  // reuse buffer as exp value
  atomicAdd(&recvSum[s1], ev);
}

// ---- Kernel 7: message, edge_out, node_out scatter (64 threads/edge) ---------
__global__ __launch_bounds__(256) void edge_final(
    const int* __restrict__ ei0, const int* __restrict__ ei1,
    const int* __restrict__ rev_idx, const int* __restrict__ aux,
    const float* __restrict__ expv, const float* __restrict__ recvSum,
    const int* __restrict__ nidx, const float* __restrict__ tabs,
    const float* __restrict__ emb_edge, float* __restrict__ node_out,
    float* __restrict__ edge_out, int eN) {
  int gid = blockIdx.x * blockDim.x + threadIdx.x;
  int e = gid >> 6, h = gid & 63;
  if (e >= eN) return;
  int s0 = ei0[e], s1 = ei1[e];
  int a = aux[e];
  int eidx = a & 15, ekidx_unused = (a >> 4) & 31, sidx = (a >> 9) & 3;
  (void)ekidx_unused;
  int eidx_r = aux[rev_idx[e]] & 15;
  float alpha = expv[e] / recvSum[s1];
  const float* Vt = tabs + 4096 + nidx[s0] * 64;
  const float* T1 = tabs + 8192 + eidx * 64;
  const float* T2 = tabs + 9216 + eidx_r * 64;
  const float* T3 = tabs + 10240 + sidx * 64;
  float msg = (Vt[h] + T1[h] + T2[h] + T3[h]) * alpha;
  edge_out[(size_t)e * HDIM + h] = emb_edge[eidx * 64 + h] + msg;
  atomicAdd(&node_out[(size_t)s1 * HDIM + h], msg);
}

extern "C" void kernel_launch(void* const* d_in, const int* in_sizes, int n_in,
                              void* d_out, int out_size, void* d_ws, size_t ws_size,
                              hipStream_t stream) {
  (void)n_in; (void)out_size; (void)ws_size;
  const int* node_states = (const int*)d_in[0];
  const int* edge_states = (const int*)d_in[1];
  const float* scalars   = (const float*)d_in[2];
  const int* edge_index  = (const int*)d_in[3];
  const int* batch_vec   = (const int*)d_in[4];
  const int* rev_idx     = (const int*)d_in[5];
  const float* emb_edge  = (const float*)d_in[6];
  const float* emb_sbv   = (const float*)d_in[7];
  const float* emb_sbr   = (const float*)d_in[8];
  const float* emb_static= (const float*)d_in[9];
  const float* Wq  = (const float*)d_in[10];
  const float* Wk  = (const float*)d_in[11];
  const float* Wv  = (const float*)d_in[12];
  const float* Wek = (const float*)d_in[13];
  // d_in[14] = Wev: unused by the reference forward pass
  const float* Wc  = (const float*)d_in[15];
  const int* tstep = (const int*)d_in[16];

  const int N = in_sizes[4];  // batch_vec
  const int E = in_sizes[5];  // rev_idx
  const int* ei0 = edge_index;
  const int* ei1 = edge_index + E;

  // workspace carve-up (256B aligned)
  char* ws = (char*)d_ws;
  size_t off = 0;
  auto alloc = [&](size_t bytes) {
    void* p = ws + off;
    off = (off + bytes + 255) & ~(size_t)255;
    return p;
  };
  unsigned long long* nodeKey = (unsigned long long*)alloc(2048ull * 8);
  unsigned long long* edgeKey = (unsigned long long*)alloc((size_t)16 * N * 8);
  float*    tabs    = (float*)alloc(11264ull * 4);
  int*      nidx    = (int*)alloc((size_t)N * 4);
  int*      aux     = (int*)alloc((size_t)E * 4);
  float*    logit   = (float*)alloc((size_t)E * 4);
  unsigned* recvMax = (unsigned*)alloc((size_t)N * 4);
  float*    recvSum = (float*)alloc((size_t)N * 4);

  float* node_out = (float*)d_out;
  float* edge_out = node_out + (size_t)N * HDIM;

  hipMemsetAsync(nodeKey, 0xFF, 2048ull * 8, stream);
  hipMemsetAsync(edgeKey, 0xFF, (size_t)16 * N * 8, stream);
  hipMemsetAsync(recvMax, 0, (size_t)N * 4, stream);
  hipMemsetAsync(recvSum, 0, (size_t)N * 4, stream);

  tables_wmma<<<44, 32, 0, stream>>>(emb_edge, emb_sbv, emb_sbr, emb_static,
                                     Wq, Wk, Wv, Wek, Wc, tabs);
  node_argmin<<<(N + 255) / 256, 256, 0, stream>>>(node_states, scalars, batch_vec,
                                                   nodeKey, N);
  node_fill<<<(int)(((size_t)N * HDIM + 255) / 256), 256, 0, stream>>>(
      node_states, batch_vec, nodeKey, emb_sbv, nidx, node_out, N);
  edge_argmin<<<(E + 255) / 256, 256, 0, stream>>>(node_states, scalars, ei0, ei1,
                                                   edgeKey, E, N);
  edge_logits<<<(int)(((size_t)E * 32 + 255) / 256), 256, 0, stream>>>(
      node_states, edge_states, scalars, ei0, ei1, edgeKey, nidx, tabs,
      aux, logit, recvMax, E, N);
  edge_exp<<<(E + 255) / 256, 256, 0, stream>>>(ei1, logit, recvMax, recvSum, tstep, E);
  edge_final<<<(int)(((size_t)E * HDIM + 255) / 256), 256, 0, stream>>>(
      ei0, ei1, rev_idx, aux, logit, recvSum, nidx, tabs, emb_edge,
      node_out, edge_out, E);
}